// SkipGramModel_86586540687485
// MI455X (gfx1250) — compile-verified
//
#include <hip/hip_runtime.h>
#include <math.h>

typedef __attribute__((ext_vector_type(2))) float v2f;
typedef __attribute__((ext_vector_type(8))) float v8f;

#define DIM 128
#define N_NEG 10
#define NVEC (1 + N_NEG)              // pos + 10 negs
#define BATCH 65536
#define TILES (BATCH / 16)            // 4096 tiles of 16 batch rows
#define WAVES_PER_BLOCK 8             // 256 threads = 8 wave32
#define NBLOCKS (TILES / WAVES_PER_BLOCK)  // 512

// One wave32 handles 16 batch elements. D = U(16x128) x V^T(128x16) via
// 32 chained V_WMMA_F32_16X16X4_F32 per vector; diagonal = the 16 dots.
// Software-pipelined: K-step k+4's 12 fragments are loaded into distinct
// live registers while the 11 WMMAs for step k execute, so the scheduler
// can use partial s_wait_loadcnt instead of serializing on loadcnt==0.
__global__ __launch_bounds__(256)
void sg_main_kernel(const int* __restrict__ pos_u,
                    const int* __restrict__ pos_v,
                    const int* __restrict__ neg_v,
                    const float* __restrict__ u_weight,
                    const float* __restrict__ v_weight,
                    float* __restrict__ partial)
{
    const int lane = threadIdx.x & 31;
    const int wave = threadIdx.x >> 5;
    const int tile = blockIdx.x * WAVES_PER_BLOCK + wave;
    const int row  = lane & 15;       // A row / B column this lane feeds
    const int hi   = lane >> 4;       // 0: K=k..k+1, 1: K=k+2..k+3 (f32 layout)
    const int b    = tile * 16 + row;

    // Per-lane row base pointers, pre-shifted by the lane's K sub-offset so
    // every load in the unrolled loop uses a constant immediate offset.
    const float* uP = u_weight + (size_t)pos_u[b] * DIM + 2 * hi;
    const float* vP[NVEC];
    vP[0] = v_weight + (size_t)pos_v[b] * DIM + 2 * hi;
#pragma unroll
    for (int n = 0; n < N_NEG; ++n)
        vP[1 + n] = v_weight + (size_t)neg_v[b * N_NEG + n] * DIM + 2 * hi;

    v8f acc[NVEC];
#pragma unroll
    for (int j = 0; j < NVEC; ++j)
#pragma unroll
        for (int e = 0; e < 8; ++e) acc[j][e] = 0.0f;

    // ---- pipeline prologue: fragments for k = 0 ----
    v2f a_cur = *(const v2f*)(uP);
    v2f b_cur[NVEC];
#pragma unroll
    for (int j = 0; j < NVEC; ++j) b_cur[j] = *(const v2f*)(vP[j]);

    // ---- steady state: prefetch k+4 while multiplying k ----
#pragma unroll
    for (int k = 0; k < DIM - 4; k += 4) {
        v2f a_nxt = *(const v2f*)(uP + k + 4);
        v2f b_nxt[NVEC];
#pragma unroll
        for (int j = 0; j < NVEC; ++j)
            b_nxt[j] = *(const v2f*)(vP[j] + k + 4);

#pragma unroll
        for (int j = 0; j < NVEC; ++j)
            acc[j] = __builtin_amdgcn_wmma_f32_16x16x4_f32(
                false, a_cur, false, b_cur[j], (short)0, acc[j], false, false);

        a_cur = a_nxt;
#pragma unroll
        for (int j = 0; j < NVEC; ++j) b_cur[j] = b_nxt[j];
    }

    // ---- epilogue: last K step (k = 124) ----
#pragma unroll
    for (int j = 0; j < NVEC; ++j)
        acc[j] = __builtin_amdgcn_wmma_f32_16x16x4_f32(
            false, a_cur, false, b_cur[j], (short)0, acc[j], false, false);

    // Diagonal extraction from C/D layout:
    //  VGPR r: lanes 0-15 -> (M=r, N=lane); lanes 16-31 -> (M=8+r, N=lane-16)
    //  diag b<8  at lane b     element b
    //  diag b>=8 at lane b+16  element b-8   (lanes 24..31, element lane-24)
    const bool hasDiag = (lane < 8) || (lane >= 24);
    const int  r = (lane < 8) ? lane : (lane - 24);

    float diag[NVEC];
#pragma unroll
    for (int j = 0; j < NVEC; ++j) {
        float d = 0.0f;
#pragma unroll
        for (int e = 0; e < 8; ++e) d += (e == r) ? acc[j][e] : 0.0f;
        diag[j] = d;
    }

    // loss: -log_sigmoid(clip(pos,±10)) + sum_n log_sigmoid(-neg_n)
    float s = fminf(fmaxf(diag[0], -10.0f), 10.0f);
    float contrib = log1pf(expf(-s));        // safe: s >= -10 -> exp <= e^10
#pragma unroll
    for (int n = 0; n < N_NEG; ++n) {
        float x = diag[1 + n];
        // log_sigmoid(-x) = min(-x,0) - log1p(exp(-|x|))  (numerically stable)
        contrib += fminf(-x, 0.0f) - log1pf(expf(-fabsf(x)));
    }
    contrib = hasDiag ? contrib : 0.0f;

    // wave32 reduction (16 live values)
#pragma unroll
    for (int m = 16; m >= 1; m >>= 1)
        contrib += __shfl_xor(contrib, m, 32);

    __shared__ float wsum[WAVES_PER_BLOCK];
    if (lane == 0) wsum[wave] = contrib;
    __syncthreads();
    if (threadIdx.x == 0) {
        float t = 0.0f;
#pragma unroll
        for (int w = 0; w < WAVES_PER_BLOCK; ++w) t += wsum[w];
        partial[blockIdx.x] = t;   // overwrites poisoned d_ws slot
    }
}

// Deterministic final reduction: single block sums 512 block partials.
__global__ __launch_bounds__(512)
void sg_reduce_kernel(const float* __restrict__ partial, float* __restrict__ out)
{
    __shared__ float red[512];
    red[threadIdx.x] = partial[threadIdx.x];
    __syncthreads();
#pragma unroll
    for (int s = 256; s > 0; s >>= 1) {
        if (threadIdx.x < s) red[threadIdx.x] += red[threadIdx.x + s];
        __syncthreads();
    }
    if (threadIdx.x == 0) out[0] = red[0] * (1.0f / (float)BATCH);
}

extern "C" void kernel_launch(void* const* d_in, const int* in_sizes, int n_in,
                              void* d_out, int out_size, void* d_ws, size_t ws_size,
                              hipStream_t stream)
{
    (void)in_sizes; (void)n_in; (void)out_size; (void)ws_size;
    const int*   pos_u    = (const int*)d_in[0];
    const int*   pos_v    = (const int*)d_in[1];
    const int*   neg_v    = (const int*)d_in[2];
    const float* u_weight = (const float*)d_in[3];
    const float* v_weight = (const float*)d_in[4];
    float* partial = (float*)d_ws;           // 512 floats of scratch
    float* out     = (float*)d_out;

    sg_main_kernel<<<NBLOCKS, 256, 0, stream>>>(pos_u, pos_v, neg_v,
                                                u_weight, v_weight, partial);
    sg_reduce_kernel<<<1, 512, 0, stream>>>(partial, out);
}